// MultiLabelSoftMarginLoss_88828513616443
// MI455X (gfx1250) — compile-verified
//
#include <hip/hip_runtime.h>
#include <hip/hip_bf16.h>
#include <cstdint>
#include <cstddef>

typedef __attribute__((ext_vector_type(2))) float v2f;
typedef __attribute__((ext_vector_type(4))) float f4;
typedef __attribute__((ext_vector_type(8))) float v8f;

#define BLK 256
#define KLBL 20

#define LOG2E 1.4426950408889634f
#define LN2   0.6931471805599453f

// t(x) = log1p(exp(-|x|)) via raw v_exp_f32 / v_log_f32.
// Safe: exp2 arg <= 0 (no overflow; underflow->0 is exact for log1p),
// log arg in [1,2] where v_log_f32 is ~1 ulp. No OCML fixup code.
__device__ __forceinline__ float lse_t(float x) {
    const float e = __builtin_amdgcn_exp2f(-LOG2E * fabsf(x));
    return LN2 * __builtin_amdgcn_logf(1.0f + e);
}

// acc += log_sigmoid(-x) = -max(x,0) - t(x)
__device__ __forceinline__ void acc_negls(float x, float& a) {
    const float e  = __builtin_amdgcn_exp2f(-LOG2E * fabsf(x));
    const float lg = __builtin_amdgcn_logf(1.0f + e);
    a -= fmaxf(x, 0.0f);
    a = __builtin_fmaf(-LN2, lg, a);
}

__device__ __forceinline__ void acc_vec4(const f4 x, float& a) {
    acc_negls(x.x, a);
    acc_negls(x.y, a);
    acc_negls(x.z, a);
    acc_negls(x.w, a);
}

// One block per row: stream the row with NT b128 loads, accumulate
// sum of log_sigmoid(-x); handle the 20 labels (gather + dedup) in LDS;
// reduce the 256 per-thread partials with V_WMMA_F32_16X16X4_F32.
__global__ __launch_bounds__(BLK)
void mlsm_row_kernel(const float* __restrict__ inp,
                     const int*   __restrict__ tgt,
                     float*       __restrict__ row_out,
                     int C)
{
    const int row = blockIdx.x;
    const int tid = threadIdx.x;
    const float* __restrict__ p = inp + (size_t)row * (size_t)C;

    __shared__ int   s_idx[KLBL];
    __shared__ float s_pos[KLBL];
    __shared__ float s_corr[KLBL];
    __shared__ int   s_uniq[KLBL];
    __shared__ float s_wave[16];   // 8 waves x {lanes 0..15, lanes 16..31}

    if (tid < KLBL) s_idx[tid] = tgt[row * KLBL + tid];
    __syncthreads();

    // Label gather: positive term for every occurrence; negative-correction
    // + unique-count only for the first occurrence of each index.
    if (tid < KLBL) {
        const int t = s_idx[tid];
        const float x = p[t];
        const float l = lse_t(x);
        s_pos[tid] = fminf(x, 0.0f) - l;
        bool dup = false;
        #pragma unroll
        for (int i = 0; i < KLBL; ++i)
            if (i < tid) dup |= (s_idx[i] == t);
        s_corr[tid] = dup ? 0.0f : (-fmaxf(x, 0.0f) - l);
        s_uniq[tid] = dup ? 0 : 1;
    }

    // ---- streaming row sum of log_sigmoid(-x) ----
    // Row base is misaligned by (row*C) mod 4 floats; peel a head so the
    // vector body uses 16B-aligned global_load_b128 (nontemporal).
    const int mis  = (int)(((size_t)row * (size_t)C) & 3);
    const int head = (4 - mis) & 3;
    const int nvec = (C - head) >> 2;
    const int tail = C - head - (nvec << 2);

    float acc0 = 0.0f, acc1 = 0.0f;
    if (tid < head) acc_negls(p[tid], acc0);

    const f4* __restrict__ vp = (const f4*)(p + head);
    int i = tid;
    for (; i + BLK < nvec; i += 2 * BLK) {
        const f4 x0 = __builtin_nontemporal_load(vp + i);
        const f4 x1 = __builtin_nontemporal_load(vp + i + BLK);
        acc_vec4(x0, acc0);
        acc_vec4(x1, acc1);
    }
    if (i < nvec) {
        const f4 x0 = __builtin_nontemporal_load(vp + i);
        acc_vec4(x0, acc0);
    }
    if (tid < tail) acc_negls(p[head + (nvec << 2) + tid], acc0);

    float acc = acc0 + acc1;

    // ---- intra-wave reduction via WMMA f32 16x16x4 ----
    // A[m,0]=acc[lane m], A[m,2]=acc[lane m+16], A[m,1]=A[m,3]=0; B=ones.
    // D[m,n] = acc[m] + acc[m+16] for all n. Lane n<16 sums D[0..7,n],
    // lane n>=16 sums D[8..15,n-16]; total wave sum = s(lane0)+s(lane16).
    v2f a; a.x = acc;  a.y = 0.0f;
    v2f b; b.x = 1.0f; b.y = 1.0f;
    v8f c = {};
    v8f d = __builtin_amdgcn_wmma_f32_16x16x4_f32(
        /*neg_a=*/false, a, /*neg_b=*/false, b,
        /*c_mod=*/(short)0, c, /*reuse_a=*/false, /*reuse_b=*/false);
    float s = d[0] + d[1] + d[2] + d[3] + d[4] + d[5] + d[6] + d[7];

    const int lane = tid & 31;
    const int wave = tid >> 5;
    if ((lane & 15) == 0) s_wave[wave * 2 + (lane >> 4)] = s;
    __syncthreads();

    // Fixed-order serial finish (deterministic, tiny).
    if (tid == 0) {
        float rowsum = 0.0f;
        #pragma unroll
        for (int i2 = 0; i2 < 16; ++i2) rowsum += s_wave[i2];
        float possum = 0.0f, corr = 0.0f; int uniq = 0;
        #pragma unroll
        for (int j = 0; j < KLBL; ++j) {
            possum += s_pos[j];
            corr   += s_corr[j];
            uniq   += s_uniq[j];
        }
        const float neg_mean = (rowsum - corr) / (float)(C - uniq);
        row_out[row] = possum * (1.0f / (float)KLBL) + neg_mean;
    }
}

// Deterministic 2048 -> 1 reduction; applies -mean.
__global__ __launch_bounds__(BLK)
void mlsm_final_kernel(const float* __restrict__ rows, float* __restrict__ out, int B)
{
    __shared__ float red[BLK];
    const int tid = threadIdx.x;
    float s = 0.0f;
    for (int i = tid; i < B; i += BLK) s += rows[i];
    red[tid] = s;
    __syncthreads();
    for (int off = BLK / 2; off > 0; off >>= 1) {
        if (tid < off) red[tid] += red[tid + off];
        __syncthreads();
    }
    if (tid == 0) out[0] = -red[0] / (float)B;
}

extern "C" void kernel_launch(void* const* d_in, const int* in_sizes, int n_in,
                              void* d_out, int out_size, void* d_ws, size_t ws_size,
                              hipStream_t stream)
{
    const float* inp = (const float*)d_in[0];
    const int*   tgt = (const int*)d_in[1];
    float*       out = (float*)d_out;
    float*       ws  = (float*)d_ws;

    const int B = in_sizes[1] / KLBL;      // 2048
    const int C = in_sizes[0] / B;         // 50257

    mlsm_row_kernel<<<B, BLK, 0, stream>>>(inp, tgt, ws, C);
    mlsm_final_kernel<<<1, BLK, 0, stream>>>(ws, out, B);
}